// ConvQuadInterp3d_22797686408190
// MI455X (gfx1250) — compile-verified
//
#include <hip/hip_runtime.h>
#include <cstdint>

// Problem shape (from setup_inputs): x : (B=2, C=8, D=8, H=256, W=256) float32
#define B_ 2
#define C_ 8
#define D_ 8
#define H_ 256
#define W_ 256

// Tile: full depth, 8 rows, 32 cols per 256-thread block.
#define TX 32
#define TY 8
constexpr int LW = TX + 2;                         // 34 floats per LDS row
constexpr int LH = TY + 2;                         // 10 rows
constexpr size_t DHW = (size_t)D_ * H_ * W_;       // 524288 per (b,c) slice
constexpr size_t COORDS_TOTAL = (size_t)B_ * C_ * 3 * DHW;  // 25165824

__global__ __launch_bounds__(256)
void ConvQuadInterp3d_kernel(const float* __restrict__ x, float* __restrict__ out) {
    // 10.9 KB LDS tile: whole depth + replicate halo in y/x.
    __shared__ float tile[D_][LH][LW];

    const int bc  = blockIdx.z;            // fused (b*C + c), 0..15
    const int y0  = blockIdx.y * TY;
    const int x0  = blockIdx.x * TX;
    const int tid = threadIdx.x;

    const float* slice = x + (size_t)bc * DHW;

    // ------------------------------------------------------------------
    // Stage haloed tile into LDS with CDNA5 async global->LDS loads.
    // Halo addresses are replicate-clamped to the array bounds so the
    // compute phase never needs special edge handling (matches edge-pad).
    // ------------------------------------------------------------------
    constexpr int NELEM = D_ * LH * LW;    // 2720 elements
    for (int i = tid; i < NELEM; i += 256) {
        int lx = i % LW;
        int r  = i / LW;
        int ly = r % LH;
        int z  = r / LH;
        int gy = y0 - 1 + ly; gy = gy < 0 ? 0 : (gy > H_ - 1 ? H_ - 1 : gy);
        int gx = x0 - 1 + lx; gx = gx < 0 ? 0 : (gx > W_ - 1 ? W_ - 1 : gx);
        unsigned voff  = (unsigned)(((z * H_ + gy) * W_ + gx) * 4);
        // Low 32 bits of the generic pointer to an LDS object == LDS offset.
        unsigned laddr = (unsigned)(unsigned long long)(void*)&tile[z][ly][lx];
        asm volatile("global_load_async_to_lds_b32 %0, %1, %2"
                     :: "v"(laddr), "v"(voff), "s"(slice)
                     : "memory");
    }
#if __has_builtin(__builtin_amdgcn_s_wait_asynccnt)
    __builtin_amdgcn_s_wait_asynccnt(0);
#else
    asm volatile("s_wait_asynccnt 0" ::: "memory");
#endif
    __syncthreads();

    // ------------------------------------------------------------------
    // Compute: one (y,x) column per thread, loop over all 8 z planes.
    // ------------------------------------------------------------------
    const int lx = tid & (TX - 1);
    const int ly = tid >> 5;
    const int gxo = x0 + lx;
    const int gyo = y0 + ly;

    float* coordZ = out + (size_t)(bc * 3 + 0) * DHW;
    float* coordX = out + (size_t)(bc * 3 + 1) * DHW;
    float* coordY = out + (size_t)(bc * 3 + 2) * DHW;
    float* ymax   = out + COORDS_TOTAL + (size_t)bc * DHW;

#pragma unroll
    for (int z = 0; z < D_; ++z) {
        const int zm = (z - 1 < 0) ? 0 : z - 1;          // replicate in z
        const int zp = (z + 1 > D_ - 1) ? D_ - 1 : z + 1;
        const int zs[3] = {zm, z, zp};

        float n[3][3][3];
#pragma unroll
        for (int a = 0; a < 3; ++a)
#pragma unroll
            for (int b = 0; b < 3; ++b)
#pragma unroll
                for (int c = 0; c < 3; ++c)
                    n[a][b][c] = tile[zs[a]][ly + b][lx + c];

        const float cc  = n[1][1][1];
        const float gxv = 0.5f * (n[1][1][2] - n[1][1][0]);
        const float gyv = 0.5f * (n[1][2][1] - n[1][0][1]);
        const float gsv = 0.5f * (n[2][1][1] - n[0][1][1]);
        const float dxx = n[1][1][0] + n[1][1][2] - 2.0f * cc;
        const float dyy = n[1][0][1] + n[1][2][1] - 2.0f * cc;
        const float dss = n[0][1][1] + n[2][1][1] - 2.0f * cc;
        const float dxy = 0.25f * (n[1][0][0] + n[1][2][2] - n[1][0][2] - n[1][2][0]);
        const float dys = 0.25f * (n[0][0][1] + n[2][2][1] - n[0][2][1] - n[2][0][1]);
        const float dxs = 0.25f * (n[0][1][0] + n[2][1][2] - n[0][1][2] - n[2][1][0]);

        // NMS: x > max(neigh, 0)  ==  (x > 0) && (x > every neighbor).
        // Conjunction of compares also matches JAX's NaN propagation.
        bool nms = cc > 0.0f;
#pragma unroll
        for (int a = 0; a < 3; ++a)
#pragma unroll
            for (int b = 0; b < 3; ++b)
#pragma unroll
                for (int c = 0; c < 3; ++c)
                    if (!(a == 1 && b == 1 && c == 1))
                        nms = nms && (cc > n[a][b][c]);

        // det(H) via the same cofactor expression as the reference.
        const float A00 = dyy * dss - dys * dys;
        const float A01 = dxs * dys - dxy * dss;
        const float A02 = dxy * dys - dyy * dxs;
        const float det = dxx * A00 + dxy * A01 + dxs * A02;

        const bool valid = (__builtin_fabsf(det) <= 3.402823466e38f) && (det != 0.0f);
        const bool good  = nms && valid;

        float d0 = 0.0f, d1 = 0.0f, d2 = 0.0f;
        if (good) {
            // Symmetric 3x3 solve via adjugate (Cramer): sol = adj(H)*b / det.
            const float inv = 1.0f / det;
            const float M11 = dxx * dss - dxs * dxs;
            const float M12 = dxy * dxs - dxx * dys;
            const float M22 = dxx * dyy - dxy * dxy;
            const float s0  = (A00 * gxv + A01 * gyv + A02 * gsv) * inv;
            const float s1  = (A01 * gxv + M11 * gyv + M12 * gsv) * inv;
            const float s2  = (A02 * gxv + M12 * gyv + M22 * gsv) * inv;
            d0 = -s0; d1 = -s1; d2 = -s2;
            const bool too_far = (__builtin_fabsf(d0) > 0.7f) ||
                                 (__builtin_fabsf(d1) > 0.7f) ||
                                 (__builtin_fabsf(d2) > 0.7f);
            if (too_far) { d0 = 0.0f; d1 = 0.0f; d2 = 0.0f; }
        }

        const float dyv = 0.5f * (gxv * d0 + gyv * d1 + gsv * d2);
        const float yv  = cc + dyv + (good ? 10.0f : 0.0f);

        const size_t idx = ((size_t)z * H_ + gyo) * W_ + gxo;
        // Streaming outputs: non-temporal so 134 MB of writes don't evict the
        // 33.5 MB input working set from L2 (input fits entirely in 192 MB L2).
        __builtin_nontemporal_store((float)z   + d2, coordZ + idx);  // grid ch0: D + dx_s
        __builtin_nontemporal_store((float)gxo + d0, coordX + idx);  // grid ch1: W + dx_x
        __builtin_nontemporal_store((float)gyo + d1, coordY + idx);  // grid ch2: H + dx_y
        __builtin_nontemporal_store(yv, ymax + idx);
    }
}

extern "C" void kernel_launch(void* const* d_in, const int* in_sizes, int n_in,
                              void* d_out, int out_size, void* d_ws, size_t ws_size,
                              hipStream_t stream) {
    (void)in_sizes; (void)n_in; (void)out_size; (void)d_ws; (void)ws_size;
    const float* x = (const float*)d_in[0];
    float* out = (float*)d_out;
    dim3 grid(W_ / TX, H_ / TY, B_ * C_);   // (8, 32, 16) blocks
    ConvQuadInterp3d_kernel<<<grid, dim3(256), 0, stream>>>(x, out);
}